// GPTRewardModel_55061480735207
// MI455X (gfx1250) — compile-verified
//
#include <hip/hip_runtime.h>

// GPT reward-model forward for MI455X (gfx1250, wave32).
//
// Shapes (fixed by reference setup_inputs): B=4 pairs, S=2048, D=4096.
//   hidden_states: (2B, S, D) fp32   = 256 MB  (streamed once -> non-temporal)
//   v_head_w:      (D,)       fp32   = 16 KB   (cached)
//   input_ids:     (2B, S)    int32  (JAX default x64-off -> int32)
//
// rewards = einsum('bsd,d->bs') is bandwidth-bound (0.5 flop/byte; ~11us
// floor at 23.3 TB/s), so precision stays fp32 and the dot products run
// through V_WMMA_F32_16X16X4_F32 (the only fp32 WMMA).
//
// Key trick: the WMMA computes C[m,n] = sum over K *slots* of A_slot*B_slot,
// so any permutation of K-slot assignment is correct as long as A and B use
// the same one. That lets each lane fetch a full float4 of its row with one
// b128 load: lane m supplies row[k..k+3], lane m+16 supplies row[k+4..k+7];
// WMMA#0 eats {k,k+1 | k+4,k+5}, WMMA#1 eats {k+2,k+3 | k+6,k+7}, with
// matching v_head_w quads as the (N-broadcast) B fragment.
//
// Parallelism: 16384/16 = 1024 tiles is too few waves to cover the HBM
// stream, so K is split into KSEG=4 segments (4096 waves); fp32 partials go
// to d_ws and are combined in fixed order (bit-deterministic, no atomics).

typedef __attribute__((ext_vector_type(2))) float v2f;
typedef __attribute__((ext_vector_type(4))) float v4f;
typedef __attribute__((ext_vector_type(8))) float v8f;

#define PAD_ID 50256

constexpr int kB    = 4;              // pairs
constexpr int kS    = 2048;           // sequence length
constexpr int kD    = 4096;           // hidden dim
constexpr int kRows = 2 * kB * kS;    // 16384 total (b,s) rows

// ---------------------------------------------------------------------------
// Kernel 1: partial[seg, row] = dot(hs[row, seg*KLEN:(seg+1)*KLEN], w[...]).
// One wave32 per (16-row tile, K segment). 8 waves / 256-thread block.
// ---------------------------------------------------------------------------
template <int KSEG>
__global__ __launch_bounds__(256) void reward_gemv_wmma(
    const float* __restrict__ hs,
    const float* __restrict__ w,
    float* __restrict__ part) {
  const int lane = threadIdx.x & 31;
  const int wid  = blockIdx.x * (blockDim.x >> 5) + (threadIdx.x >> 5);
  const int tile = wid / KSEG;
  const int seg  = wid % KSEG;
  const int row0 = tile * 16;

  // ISA 32-bit A 16x4 layout: lanes 0-15 carry two K slots in VGPR0/1,
  // lanes 16-31 carry the other two. M = lane % 16 (row appears in both
  // halves). We assign this lane the K quad starting at 4*half of each
  // 8-wide chunk; B uses the identical slotting so the mapping cancels.
  const int half = lane >> 4;
  const int m    = lane & 15;

  constexpr int KLEN = kD / KSEG;
  const int k0 = seg * KLEN;

  const float* __restrict__ rowp = hs + (size_t)(row0 + m) * kD + k0 + 4 * half;
  const float* __restrict__ wp   = w + k0 + 4 * half;

  v8f acc0 = {};
  v8f acc1 = {};

#pragma unroll 4
  for (int k = 0; k < KLEN; k += 8) {
    // One b128 per lane feeds two WMMAs. hidden_states is streamed once
    // (256 MB > 192 MB L2) -> non-temporal; w stays cache-resident.
    v4f a  = __builtin_nontemporal_load((const v4f*)(rowp + k));
    v4f bw = *(const v4f*)(wp + k);

    v2f a0 = {a.x, a.y};
    v2f b0 = {bw.x, bw.y};
    acc0 = __builtin_amdgcn_wmma_f32_16x16x4_f32(
        /*neg_a=*/false, a0, /*neg_b=*/false, b0,
        /*c_mod=*/(short)0, acc0, /*reuse_a=*/false, /*reuse_b=*/false);

    v2f a1 = {a.z, a.w};
    v2f b1 = {bw.z, bw.w};
    acc1 = __builtin_amdgcn_wmma_f32_16x16x4_f32(
        false, a1, false, b1, (short)0, acc1, false, false);
  }

  v8f acc = acc0 + acc1;

  // C/D layout: VGPR r of lanes 0-15 holds M=r, lanes 16-31 hold M=8+r;
  // every N column is identical (B was N-broadcast). Lane 0 writes rows
  // row0..row0+7, lane 16 writes row0+8..row0+15.
  if (m == 0) {
    float* outp = part + (size_t)seg * kRows + row0 + 8 * half;
#pragma unroll
    for (int r = 0; r < 8; ++r) outp[r] = acc[r];
  }
}

// ---------------------------------------------------------------------------
// Kernel 1b: rewards[i] = sum_seg part[seg][i]  (fixed order -> deterministic)
// ---------------------------------------------------------------------------
template <int KSEG>
__global__ __launch_bounds__(256) void combine_partials(
    const float* __restrict__ part,
    float* __restrict__ rewards) {
  const int i = blockIdx.x * blockDim.x + threadIdx.x;
  if (i < kRows) {
    float s = 0.f;
#pragma unroll
    for (int g = 0; g < KSEG; ++g) s += part[(size_t)g * kRows + i];
    rewards[i] = s;
  }
}

// ---------------------------------------------------------------------------
// Kernel 2: per-pair indices, masked means, end scores, loss.
// Single 256-thread block; deterministic LDS tree reductions (no atomics).
// ---------------------------------------------------------------------------
__device__ __forceinline__ float blk_sum(float v, float* buf) {
  const int t = threadIdx.x;
  buf[t] = v;
  __syncthreads();
  for (int o = 128; o > 0; o >>= 1) {
    if (t < o) buf[t] += buf[t + o];
    __syncthreads();
  }
  float r = buf[0];
  __syncthreads();
  return r;
}

__device__ __forceinline__ int blk_min(int v, int* buf) {
  const int t = threadIdx.x;
  buf[t] = v;
  __syncthreads();
  for (int o = 128; o > 0; o >>= 1) {
    if (t < o) buf[t] = min(buf[t], buf[t + o]);
    __syncthreads();
  }
  int r = buf[0];
  __syncthreads();
  return r;
}

__global__ __launch_bounds__(256) void pair_reduce(
    const int* __restrict__ ids,
    const float* __restrict__ rewards,
    float* __restrict__ out) {
  __shared__ float fbuf[256];
  __shared__ int   ibuf[256];
  const int t = threadIdx.x;

  float loss_acc = 0.f;

  for (int b = 0; b < kB; ++b) {
    const int*   ch = ids + (size_t)b * kS;
    const int*   rj = ids + (size_t)(kB + b) * kS;
    const float* cr = rewards + (size_t)b * kS;
    const float* rr = rewards + (size_t)(kB + b) * kS;

    // First-index scans (min index satisfying each predicate, default S).
    int ldiv = kS, lc = kS, lr = kS;
    for (int s = t; s < kS; s += 256) {
      int c = ch[s], r = rj[s];
      if (c != r)      ldiv = min(ldiv, s);
      if (c == PAD_ID) lc   = min(lc, s);
      if (r == PAD_ID) lr   = min(lr, s);
    }
    const int div = blk_min(ldiv, ibuf);
    const int cin = blk_min(lc, ibuf);
    const int rin = blk_min(lr, ibuf);
    const int end = max(cin, rin);

    // Mask [div, end) is contiguous -> cnt needs no reduction.
    const int   cnt  = max(end - div, 0);
    const float fcnt = (float)max(cnt, 1);

    float sc = 0.f, sr = 0.f, sl = 0.f;
    for (int s = t; s < kS; s += 256) {
      if (s >= div && s < end) {
        float c = cr[s], r = rr[s];
        sc += c;
        sr += r;
        // -log_sigmoid(c - r) = softplus(r - c), numerically stable form.
        float z = r - c;
        sl += fmaxf(z, 0.f) + log1pf(expf(-fabsf(z)));
      }
    }
    sc = blk_sum(sc, fbuf);
    sr = blk_sum(sr, fbuf);
    sl = blk_sum(sl, fbuf);

    loss_acc += sl / fcnt;

    if (t == 0) {
      const int base = 1 + kRows;          // after loss + rewards
      out[base + b]          = sc / fcnt;  // chosen_mean_scores
      out[base + kB + b]     = sr / fcnt;  // rejected_mean_scores
      const int last = max(end - 1, 0);
      out[base + 2 * kB + b] = cr[last];   // chosen_end_scores
      out[base + 3 * kB + b] = rr[last];   // rejected_end_scores
    }
    __syncthreads();
  }

  if (t == 0) out[0] = loss_acc / (float)kB;  // loss
}

// ---------------------------------------------------------------------------
extern "C" void kernel_launch(void* const* d_in, const int* in_sizes, int n_in,
                              void* d_out, int out_size, void* d_ws, size_t ws_size,
                              hipStream_t stream) {
  (void)in_sizes; (void)n_in; (void)out_size;

  const float* hs  = (const float*)d_in[0];  // (2B, S, D) fp32
  const float* w   = (const float*)d_in[1];  // (D,) fp32
  const int*   ids = (const int*)d_in[2];    // (2B, S) int32

  float* out     = (float*)d_out;
  float* rewards = out + 1;                  // out[1 .. 1+16384)

  constexpr int KSEG = 4;
  const dim3 blk(256);

  if (ws_size >= (size_t)KSEG * kRows * sizeof(float)) {
    // K-split path: 4096 waves streaming, fixed-order combine.
    float* part = (float*)d_ws;
    const int waves = (kRows / 16) * KSEG;   // 4096 waves, 8 per block
    reward_gemv_wmma<KSEG><<<dim3(waves / 8), blk, 0, stream>>>(hs, w, part);
    combine_partials<KSEG><<<dim3((kRows + 255) / 256), blk, 0, stream>>>(part, rewards);
  } else {
    // Fallback: one wave per tile writes rewards directly.
    reward_gemv_wmma<1><<<dim3((kRows / 16) / 8), blk, 0, stream>>>(hs, w, rewards);
  }

  pair_reduce<<<1, blk, 0, stream>>>(ids, rewards, out);
}